// AALPositionalEmbedding_25975962206429
// MI455X (gfx1250) — compile-verified
//
#include <hip/hip_runtime.h>
#include <cstdint>

// Problem constants (match reference)
#define B_DIM 16
#define N_DIM 8192
#define C_DIM 768          // embed dim
#define AAL_D 91
#define AAL_H 109
#define AAL_W 91
#define REGION_MAX 116

#define NPOINTS (B_DIM * N_DIM)       // 131072
#define WAVES_PER_BLOCK 8             // 256 threads / wave32
#define PTS_PER_WAVE 4
#define V4_PER_LANE (C_DIM / 4 / 32)  // 6 x b128 per lane per row

typedef float v4f __attribute__((ext_vector_type(4)));
typedef int   v4i __attribute__((ext_vector_type(4)));

#if __has_builtin(__builtin_amdgcn_global_load_async_to_lds_b128) &&  \
    __has_builtin(__builtin_amdgcn_global_store_async_from_lds_b128) && \
    __has_builtin(__builtin_amdgcn_s_wait_asynccnt)
#define USE_ASYNC_LDS 1
#else
#define USE_ASYNC_LDS 0
#endif

#define GAS __attribute__((address_space(1)))
#define LAS __attribute__((address_space(3)))

__global__ __launch_bounds__(256) void aal_embed_kernel(
    const float* __restrict__ centers,   // [B*N*3]
    const float* __restrict__ mri_aff,   // [16] row-major 4x4
    const float* __restrict__ aal_aff,   // [16] row-major 4x4
    const float* __restrict__ embed,     // [(REGION_MAX+1)*C]
    const int*   __restrict__ aal_data,  // [D*H*W]
    float* __restrict__ out)             // [B*N*C]
{
    const int lane = threadIdx.x & 31;
    const int warp = threadIdx.x >> 5;
    const int wave_id = blockIdx.x * WAVES_PER_BLOCK + warp;
    const int base = wave_id * PTS_PER_WAVE;
    if (base >= NPOINTS) return;

    // ---- uniform affine setup (kernarg-uniform; compiler scalarizes) ----
    const float a00 = aal_aff[0],  a01 = aal_aff[1],  a02 = aal_aff[2],  ta0 = aal_aff[3];
    const float a10 = aal_aff[4],  a11 = aal_aff[5],  a12 = aal_aff[6],  ta1 = aal_aff[7];
    const float a20 = aal_aff[8],  a21 = aal_aff[9],  a22 = aal_aff[10], ta2 = aal_aff[11];

    const float det = a00 * (a11 * a22 - a12 * a21)
                    - a01 * (a10 * a22 - a12 * a20)
                    + a02 * (a10 * a21 - a11 * a20);
    const float r = 1.0f / det;
    const float i00 =  (a11 * a22 - a12 * a21) * r;
    const float i01 = -(a01 * a22 - a02 * a21) * r;
    const float i02 =  (a01 * a12 - a02 * a11) * r;
    const float i10 = -(a10 * a22 - a12 * a20) * r;
    const float i11 =  (a00 * a22 - a02 * a20) * r;
    const float i12 = -(a00 * a12 - a02 * a10) * r;
    const float i20 =  (a10 * a21 - a11 * a20) * r;
    const float i21 = -(a00 * a21 - a01 * a20) * r;
    const float i22 =  (a00 * a11 - a01 * a10) * r;

    const float m00 = mri_aff[0],  m01 = mri_aff[1],  m02 = mri_aff[2],  tm0 = mri_aff[3];
    const float m10 = mri_aff[4],  m11 = mri_aff[5],  m12 = mri_aff[6],  tm1 = mri_aff[7];
    const float m20 = mri_aff[8],  m21 = mri_aff[9],  m22 = mri_aff[10], tm2 = mri_aff[11];

    const float g00 = i00 * m00 + i01 * m10 + i02 * m20;
    const float g01 = i00 * m01 + i01 * m11 + i02 * m21;
    const float g02 = i00 * m02 + i01 * m12 + i02 * m22;
    const float g10 = i10 * m00 + i11 * m10 + i12 * m20;
    const float g11 = i10 * m01 + i11 * m11 + i12 * m21;
    const float g12 = i10 * m02 + i11 * m12 + i12 * m22;
    const float g20 = i20 * m00 + i21 * m10 + i22 * m20;
    const float g21 = i20 * m01 + i21 * m11 + i22 * m21;
    const float g22 = i20 * m02 + i21 * m12 + i22 * m22;
    const float d0 = tm0 - ta0, d1 = tm1 - ta1, d2 = tm2 - ta2;
    const float t0 = i00 * d0 + i01 * d1 + i02 * d2;
    const float t1 = i10 * d0 + i11 * d1 + i12 * d2;
    const float t2 = i20 * d0 + i21 * d1 + i22 * d2;

    // region id of point p (wave-uniform; lanes compute redundantly)
    auto region_of = [&](int p) -> int {
        const float x = centers[3 * p + 0];
        const float y = centers[3 * p + 1];
        const float z = centers[3 * p + 2];
        const float vx = g00 * x + g01 * y + g02 * z + t0;
        const float vy = g10 * x + g11 * y + g12 * z + t1;
        const float vz = g20 * x + g21 * y + g22 * z + t2;
        const int ix = (int)rintf(vx);   // jnp.round = half-to-even
        const int iy = (int)rintf(vy);
        const int iz = (int)rintf(vz);
        const bool inb = (ix >= 0) & (ix < AAL_D) &
                         (iy >= 0) & (iy < AAL_H) &
                         (iz >= 0) & (iz < AAL_W);
        const int cx = min(max(ix, 0), AAL_D - 1);
        const int cy = min(max(iy, 0), AAL_H - 1);
        const int cz = min(max(iz, 0), AAL_W - 1);
        const int region = aal_data[(cx * AAL_H + cy) * AAL_W + cz];
        const bool valid = inb && (region >= 0) && (region <= REGION_MAX);
        return valid ? region : 0;
    };

#if USE_ASYNC_LDS
    // Two 3KB staging buffers per wave; async engine moves data L2->LDS->HBM
    // without touching VGPRs. Store(k) overlaps Load(k+1) and rid(k+2) math.
    __shared__ float smem[WAVES_PER_BLOCK * 2 * C_DIM];
    v4f* const bufs[2] = {
        (v4f*)(smem + (warp * 2 + 0) * C_DIM),
        (v4f*)(smem + (warp * 2 + 1) * C_DIM)
    };

    // prime the pipeline: async-load row for point 0
    {
        const int rid0 = region_of(base);
        const v4f* srow = (const v4f*)(embed + (size_t)rid0 * C_DIM);
#pragma unroll
        for (int i = 0; i < V4_PER_LANE; ++i)
            __builtin_amdgcn_global_load_async_to_lds_b128(
                (GAS v4i*)(srow + lane + i * 32),
                (LAS v4i*)(bufs[0] + lane + i * 32), 0, 0);
    }

#pragma unroll
    for (int k = 0; k < PTS_PER_WAVE; ++k) {
        int nrid = 0;
        if (k + 1 < PTS_PER_WAVE) nrid = region_of(base + k + 1);  // overlaps in-flight DMA

        // Drain: load(k) complete (safe to store), store(k-1) complete
        // (safe to overwrite buffer (k+1)&1).
        __builtin_amdgcn_s_wait_asynccnt(0);

        if (k + 1 < PTS_PER_WAVE) {
            const v4f* srow = (const v4f*)(embed + (size_t)nrid * C_DIM);
#pragma unroll
            for (int i = 0; i < V4_PER_LANE; ++i)
                __builtin_amdgcn_global_load_async_to_lds_b128(
                    (GAS v4i*)(srow + lane + i * 32),
                    (LAS v4i*)(bufs[(k + 1) & 1] + lane + i * 32), 0, 0);
        }

        v4f* drow = (v4f*)(out + (size_t)(base + k) * C_DIM);
#pragma unroll
        for (int i = 0; i < V4_PER_LANE; ++i)
            __builtin_amdgcn_global_store_async_from_lds_b128(
                (GAS v4i*)(drow + lane + i * 32),
                (LAS v4i*)(bufs[k & 1] + lane + i * 32), 0, /*TH=NT*/1);
    }
    // S_ENDPGM performs an implicit wait-idle, draining ASYNCcnt.
#else
    // Fallback: direct VGPR copy, b128 loads + non-temporal b128 stores.
#pragma unroll
    for (int k = 0; k < PTS_PER_WAVE; ++k) {
        const int p = base + k;
        const int rid = region_of(p);
        const v4f* __restrict__ src = (const v4f*)(embed + (size_t)rid * C_DIM);
        v4f* __restrict__ dst = (v4f*)(out + (size_t)p * C_DIM);
#pragma unroll
        for (int i = 0; i < V4_PER_LANE; ++i) {
            v4f v = src[lane + i * 32];
            __builtin_nontemporal_store(v, dst + lane + i * 32);
        }
    }
#endif
}

extern "C" void kernel_launch(void* const* d_in, const int* in_sizes, int n_in,
                              void* d_out, int out_size, void* d_ws, size_t ws_size,
                              hipStream_t stream) {
    const float* centers  = (const float*)d_in[0];  // [B,N,3] f32
    const float* mri_aff  = (const float*)d_in[1];  // [4,4]   f32
    const float* aal_aff  = (const float*)d_in[2];  // [4,4]   f32
    const float* embed    = (const float*)d_in[3];  // [117,768] f32
    const int*   aal_data = (const int*)d_in[4];    // [91,109,91] i32
    float* out = (float*)d_out;                     // [B,N,768] f32

    const int blocks = NPOINTS / (WAVES_PER_BLOCK * PTS_PER_WAVE);  // 4096
    aal_embed_kernel<<<blocks, 256, 0, stream>>>(centers, mri_aff, aal_aff,
                                                 embed, aal_data, out);
}